// MiNbaseNet_644245095056
// MI455X (gfx1250) — compile-verified
//
#include <hip/hip_runtime.h>

typedef __attribute__((ext_vector_type(2))) float v2f;
typedef __attribute__((ext_vector_type(8))) float v8f;

#define NSAMP 8192
#define DDIM  768
#define BDIM  1024
#define CCLS  100

// ---------------- zero int scratch ----------------
__global__ void k_zero_ints(int* p, int n) {
  int i = blockIdx.x * blockDim.x + threadIdx.x;
  if (i < n) p[i] = 0;
}

// ---------------- feat = relu(X @ W) ----------------
// X:[8192,768] rm, W:[768,1024] rm, feat:[8192,1024] rm
// 256 threads = 8 waves; 128x128 output tile; K staged 16 at a time via LDS.
__global__ __launch_bounds__(256)
void k_gemm_relu(const float* __restrict__ X, const float* __restrict__ W,
                 float* __restrict__ feat) {
  __shared__ float Xt[16][132];   // Xt[k][m], padded so k+2 rows hit new banks
  __shared__ float Wt[16][132];   // Wt[k][n]

  const int t    = threadIdx.x;
  const int lane = t & 31;
  const int w    = t >> 5;
  const int wm   = w >> 2;        // 0..1  -> 64-row strip
  const int wn   = w & 3;         // 0..3  -> 32-col strip
  const int lo   = lane & 15;
  const int hi   = lane >> 4;
  const int blockM = blockIdx.y * 128;
  const int blockN = blockIdx.x * 128;

  v8f acc[4][2] = {};

  const int xm  = t >> 1;          // 0..127 row of X tile
  const int xk  = (t & 1) * 8;     // k sub-block 0 or 8
  const int wk  = t >> 4;          // 0..15 row of W tile
  const int wq  = (t & 15) * 8;    // col group

  for (int kt = 0; kt < DDIM / 16; ++kt) {
    const int k0 = kt * 16;
    // stage X transposed
    {
      const float* src = X + (size_t)(blockM + xm) * DDIM + k0 + xk;
      const float4 xa = *(const float4*)(src);
      const float4 xb = *(const float4*)(src + 4);
      Xt[xk + 0][xm] = xa.x; Xt[xk + 1][xm] = xa.y;
      Xt[xk + 2][xm] = xa.z; Xt[xk + 3][xm] = xa.w;
      Xt[xk + 4][xm] = xb.x; Xt[xk + 5][xm] = xb.y;
      Xt[xk + 6][xm] = xb.z; Xt[xk + 7][xm] = xb.w;
    }
    // stage W direct
    {
      const float* src = W + (size_t)(k0 + wk) * BDIM + blockN + wq;
      *(float4*)&Wt[wk][wq]     = *(const float4*)(src);
      *(float4*)&Wt[wk][wq + 4] = *(const float4*)(src + 4);
    }
    __syncthreads();

#pragma unroll
    for (int k = 0; k < 4; ++k) {
      const int kk = k * 4 + hi * 2;     // K pair per ISA 16x4 f32 layout
      v2f a[4], b[2];
#pragma unroll
      for (int mt = 0; mt < 4; ++mt) {
        const int m = wm * 64 + mt * 16 + lo;
        a[mt].x = Xt[kk][m];
        a[mt].y = Xt[kk + 1][m];
      }
#pragma unroll
      for (int nt = 0; nt < 2; ++nt) {
        const int n = wn * 32 + nt * 16 + lo;
        b[nt].x = Wt[kk][n];
        b[nt].y = Wt[kk + 1][n];
      }
#pragma unroll
      for (int mt = 0; mt < 4; ++mt)
#pragma unroll
        for (int nt = 0; nt < 2; ++nt)
          acc[mt][nt] = __builtin_amdgcn_wmma_f32_16x16x4_f32(
              false, a[mt], false, b[nt], (short)0, acc[mt][nt], false, false);
    }
    __syncthreads();
  }

  // ReLU + store (C/D layout: VGPR r, lane: M = r + 8*hi, N = lo)
#pragma unroll
  for (int mt = 0; mt < 4; ++mt)
#pragma unroll
    for (int nt = 0; nt < 2; ++nt)
#pragma unroll
      for (int r = 0; r < 8; ++r) {
        float v = acc[mt][nt][r];
        v = v > 0.f ? v : 0.f;
        const int row = blockM + wm * 64 + mt * 16 + r + 8 * hi;
        const int col = blockN + wn * 32 + nt * 16 + lo;
        feat[(size_t)row * BDIM + col] = v;
      }
}

// ---------------- label histogram ----------------
__global__ void k_hist(const int* __restrict__ labels, int* __restrict__ counts) {
  int i = blockIdx.x * blockDim.x + threadIdx.x;
  if (i < NSAMP) atomicAdd(&counts[labels[i]], 1);
}

// ---------------- prefix scan over 100 classes (trivial) ----------------
__global__ void k_scan(const int* __restrict__ counts, int* __restrict__ offsets,
                       int* __restrict__ cursors, float* __restrict__ counts_f) {
  if (threadIdx.x == 0 && blockIdx.x == 0) {
    int run = 0;
    for (int c = 0; c < CCLS; ++c) {
      offsets[c]  = run;
      cursors[c]  = run;
      counts_f[c] = (float)counts[c];
      run += counts[c];
    }
  }
}

// ---------------- scatter sample ids by class ----------------
__global__ void k_scatter(const int* __restrict__ labels, int* __restrict__ cursors,
                          int* __restrict__ idx_sorted) {
  int i = blockIdx.x * blockDim.x + threadIdx.x;
  if (i < NSAMP) {
    int pos = atomicAdd(&cursors[labels[i]], 1);
    idx_sorted[pos] = i;
  }
}

// ---------------- mu[c] = sum of feat rows of class c ----------------
__global__ __launch_bounds__(256)
void k_mu(const float* __restrict__ feat, const int* __restrict__ idx_sorted,
          const int* __restrict__ counts, const int* __restrict__ offsets,
          float* __restrict__ mu) {
  const int c = blockIdx.x;
  const int t = threadIdx.x;           // handles cols 4t..4t+3
  const int nc  = counts[c];
  const int off = offsets[c];
  float4 acc = make_float4(0.f, 0.f, 0.f, 0.f);
  for (int s = 0; s < nc; ++s) {
    const int row = idx_sorted[off + s];
    const float4 v = *(const float4*)(feat + (size_t)row * BDIM + t * 4);
    acc.x += v.x; acc.y += v.y; acc.z += v.z; acc.w += v.w;
  }
  *(float4*)(mu + (size_t)c * BDIM + t * 4) = acc;
}

// ---------------- phi[c] = F_c^T F_c, symmetric 64x64 tiles ----------------
// grid.x = 136 upper-triangle tile pairs (ti<=tj), grid.y = class;
// 128 threads = 4 waves (2x2). Mirror tile written via LDS transpose so
// both stores stay coalesced.
__global__ __launch_bounds__(128)
void k_phi(const float* __restrict__ feat, const int* __restrict__ idx_sorted,
           const int* __restrict__ counts, const int* __restrict__ offsets,
           float* __restrict__ phi) {
  __shared__ float At[16][68];    // At[sample][col in I-range], padded
  __shared__ float Bt[16][68];    // Bt[sample][col in J-range], padded
  __shared__ float Tr[64][65];    // transpose staging, conflict-free

  // map linear upper-triangle index -> (ti, tj), uniform per block
  int u = blockIdx.x;
  int ti = 0;
  while (u >= 16 - ti) { u -= 16 - ti; ++ti; }
  const int tj = ti + u;
  const int c  = blockIdx.y;
  const int I = ti * 64, J = tj * 64;

  const int t    = threadIdx.x;
  const int lane = t & 31;
  const int w    = t >> 5;
  const int wmc  = w >> 1;       // 0..1
  const int wnc  = w & 1;        // 0..1
  const int lo   = lane & 15;
  const int hi   = lane >> 4;

  const int nc  = counts[c];
  const int off = offsets[c];

  v8f acc[2][2] = {};

  const int nchunks = (nc + 15) >> 4;
  for (int ch = 0; ch < nchunks; ++ch) {
    const int sc = ch * 16;
    // stage 16 gathered rows (zero-padded past nc)
#pragma unroll
    for (int i2 = 0; i2 < 2; ++i2) {
      const int fi = t + i2 * 128;      // 0..255
      const int s  = fi >> 4;           // sample slot 0..15
      const int q  = fi & 15;           // float4 index within 64 cols
      float4 va = make_float4(0.f, 0.f, 0.f, 0.f);
      float4 vb = va;
      if (sc + s < nc) {
        const int row = idx_sorted[off + sc + s];
        va = *(const float4*)(feat + (size_t)row * BDIM + I + q * 4);
        vb = *(const float4*)(feat + (size_t)row * BDIM + J + q * 4);
      }
      *(float4*)&At[s][q * 4] = va;
      *(float4*)&Bt[s][q * 4] = vb;
    }
    __syncthreads();

#pragma unroll
    for (int k = 0; k < 4; ++k) {
      const int kk = k * 4 + hi * 2;
      v2f a[2], b[2];
#pragma unroll
      for (int mt = 0; mt < 2; ++mt) {
        const int m = wmc * 32 + mt * 16 + lo;
        a[mt].x = At[kk][m];             // A[M][K] = feat[sample K][I+M]
        a[mt].y = At[kk + 1][m];
      }
#pragma unroll
      for (int nt = 0; nt < 2; ++nt) {
        const int n = wnc * 32 + nt * 16 + lo;
        b[nt].x = Bt[kk][n];             // B[K][N] = feat[sample K][J+N]
        b[nt].y = Bt[kk + 1][n];
      }
#pragma unroll
      for (int mt = 0; mt < 2; ++mt)
#pragma unroll
        for (int nt = 0; nt < 2; ++nt)
          acc[mt][nt] = __builtin_amdgcn_wmma_f32_16x16x4_f32(
              false, a[mt], false, b[nt], (short)0, acc[mt][nt], false, false);
    }
    __syncthreads();
  }

  float* base = phi + (size_t)c * BDIM * BDIM;

  // direct tile (rows I..I+63, cols J..J+63): coalesced from registers;
  // simultaneously stash into Tr for the mirror write.
#pragma unroll
  for (int mt = 0; mt < 2; ++mt)
#pragma unroll
    for (int nt = 0; nt < 2; ++nt)
#pragma unroll
      for (int r = 0; r < 8; ++r) {
        const int lr = wmc * 32 + mt * 16 + r + 8 * hi;   // local row 0..63
        const int lc = wnc * 32 + nt * 16 + lo;           // local col 0..63
        const float v = acc[mt][nt][r];
        base[(size_t)(I + lr) * BDIM + (J + lc)] = v;
        Tr[lr][lc] = v;
      }

  if (ti != tj) {
    __syncthreads();
    // mirror tile (rows J..J+63, cols I..I+63): mirror[r][c] = Tr[c][r].
    // Each thread writes 32 elements as 8 coalesced float4 stores.
    const int mr = t >> 1;               // 0..63
    const int cb = (t & 1) * 32;         // 0 or 32
#pragma unroll
    for (int q = 0; q < 8; ++q) {
      const int mc = cb + q * 4;
      float4 v;
      v.x = Tr[mc + 0][mr];
      v.y = Tr[mc + 1][mr];
      v.z = Tr[mc + 2][mr];
      v.w = Tr[mc + 3][mr];
      *(float4*)&base[(size_t)(J + mr) * BDIM + (I + mc)] = v;
    }
  }
}

extern "C" void kernel_launch(void* const* d_in, const int* in_sizes, int n_in,
                              void* d_out, int out_size, void* d_ws, size_t ws_size,
                              hipStream_t stream) {
  const float* X      = (const float*)d_in[0];
  const float* W      = (const float*)d_in[1];
  const int*   labels = (const int*)d_in[2];

  float* out      = (float*)d_out;
  float* phi      = out;                                   // [100,1024,1024]
  float* mu       = out + (size_t)CCLS * BDIM * BDIM;      // [100,1024]
  float* counts_f = mu + (size_t)CCLS * BDIM;              // [100]

  float* feat = (float*)d_ws;                              // [8192,1024] = 32 MB
  int* ints       = (int*)((char*)d_ws + (size_t)NSAMP * BDIM * sizeof(float));
  int* counts_i   = ints;          // [128]
  int* offsets    = ints + 128;    // [128]
  int* cursors    = ints + 256;    // [128]
  int* idx_sorted = ints + 384;    // [8192]

  k_zero_ints<<<1, 256, 0, stream>>>(counts_i, 128);

  dim3 ggrid(BDIM / 128, NSAMP / 128);                     // 8 x 64
  k_gemm_relu<<<ggrid, 256, 0, stream>>>(X, W, feat);

  k_hist<<<NSAMP / 256, 256, 0, stream>>>(labels, counts_i);
  k_scan<<<1, 1, 0, stream>>>(counts_i, offsets, cursors, counts_f);
  k_scatter<<<NSAMP / 256, 256, 0, stream>>>(labels, cursors, idx_sorted);

  k_mu<<<CCLS, 256, 0, stream>>>(feat, idx_sorted, counts_i, offsets, mu);

  dim3 pgrid(136, CCLS);                 // upper-triangle 64x64 tile pairs x classes
  k_phi<<<pgrid, 128, 0, stream>>>(feat, idx_sorted, counts_i, offsets, phi);
}